// KDTreeSampleLayer_75204877353750
// MI455X (gfx1250) — compile-verified
//
#include <hip/hip_runtime.h>

// KNN (K=16) over B=4, N=32768, nqueries=2048, stride=16 query subsample.
// Distances computed as |q|^2 + |x|^2 - 2 q.x via V_WMMA_F32_16X16X4_F32:
//   A (16x4)  = -2 * query coords (x,y,z,0)   -> lanes 0-15 K={0,1}, lanes 16-31 K={2,3}
//   B (4x16)  =      point coords (x,y,z,0)   -> same K split, N = lane%16
//   C (16x16) = |q_M|^2 + |x_N|^2             -> D = full squared-distance tile
// Per-lane register top-16 lists, LDS transpose + 8-way merge with (dist,idx)
// lexicographic tie-break to match jax.lax.top_k.
//
// Sync design: the distance-staging slab DL[wave] is wave-private. CDNA5 LDS
// ops from one wave complete in order (DScnt in-order, ISA ch.15.5), so
// cross-lane store->load within the wave needs only a compiler scheduling
// fence (__builtin_amdgcn_wave_barrier), not a workgroup barrier. Real
// __syncthreads() only where waves actually exchange data (query stage, merge).

typedef float v2f __attribute__((ext_vector_type(2)));
typedef float v8f __attribute__((ext_vector_type(8)));

#define BATCH   4
#define NPTS    32768
#define NQ      2048
#define KNN     16
#define QSTRIDE 16          // NPTS / NQ
#define WAVES   4
#define TPB     (WAVES * 32)
#define PTS_PER_WAVE (NPTS / WAVES)   // 8192
#define ROWS    18          // padded LDS row stride (floats): bank-conflict-free

__global__ __launch_bounds__(TPB)
void knn_wmma_kernel(const float* __restrict__ xyz,
                     long long* __restrict__ out_idx,
                     float* __restrict__ out_pts)
{
    __shared__ float QL[16 * 4];                    // query x,y,z,|q|^2
    __shared__ float DL[WAVES][2][16 * ROWS];       // per-wave distance staging
    __shared__ float MD[16 * 8 * 16];               // merge: dists
    __shared__ int   MI[16 * 8 * 16];               // merge: indices

    const int tid  = threadIdx.x;
    const int wave = tid >> 5;
    const int lane = tid & 31;
    const int hi   = lane >> 4;     // which K-half / which 16-pt tile this lane owns
    const int n    = lane & 15;     // M for A, N for B/C/D, query id in selection

    const int b     = blockIdx.x / (NQ / 16);
    const int tile  = blockIdx.x % (NQ / 16);
    const int qbase = tile * 16;

    const float* X = xyz + (size_t)b * NPTS * 3;

    // ---- stage 16 queries (coords + |q|^2) into LDS; emit pts output ----
    if (tid < 16) {
        int qp = (qbase + tid) * QSTRIDE;
        float x = X[qp * 3 + 0];
        float y = X[qp * 3 + 1];
        float z = X[qp * 3 + 2];
        QL[tid * 4 + 0] = x;
        QL[tid * 4 + 1] = y;
        QL[tid * 4 + 2] = z;
        QL[tid * 4 + 3] = x * x + y * y + z * z;
        float* op = out_pts + ((size_t)b * NQ + qbase + tid) * 3;
        op[0] = x; op[1] = y; op[2] = z;
    }
    __syncthreads();

    // ---- A matrix registers: -2 * query, per ISA 16x4 layout ----
    v2f a;
    if (hi == 0) { a.x = -2.0f * QL[n * 4 + 0]; a.y = -2.0f * QL[n * 4 + 1]; }
    else         { a.x = -2.0f * QL[n * 4 + 2]; a.y = 0.0f; }

    float qn[8];
#pragma unroll
    for (int v = 0; v < 8; ++v) qn[v] = QL[(v + 8 * hi) * 4 + 3];

    // ---- per-lane sorted top-16 (dist asc, idx asc on ties) ----
    float bd[KNN];
    int   bi[KNN];
#pragma unroll
    for (int k = 0; k < KNN; ++k) { bd[k] = 3.0e38f; bi[k] = 0x7ffffff0; }

    const int pstart = wave * PTS_PER_WAVE;
    const int pend   = pstart + PTS_PER_WAVE;

    for (int base = pstart; base < pend; base += 32) {
        // prefetch next chunk of points into caches
        if (base + 32 < pend)
            __builtin_prefetch(&X[(size_t)(base + 32 + lane) * 3], 0, 3);

#pragma unroll
        for (int t = 0; t < 2; ++t) {
            int P = base + t * 16 + n;
            float px = X[P * 3 + 0];
            float py = X[P * 3 + 1];
            float pz = X[P * 3 + 2];
            float xn = px * px + py * py + pz * pz;

            v2f bm;
            bm.x = hi ? pz : px;        // K=0 (lo) / K=2 (hi)
            bm.y = hi ? 0.0f : py;      // K=1 (lo) / K=3 (hi)

            v8f c;
#pragma unroll
            for (int v = 0; v < 8; ++v) c[v] = qn[v] + xn;

            // D[M][N] = |q_M|^2 + |x_N|^2 - 2 q_M . x_N  == squared distance
            c = __builtin_amdgcn_wmma_f32_16x16x4_f32(
                    false, a, false, bm, (short)0, c, false, false);

            // transpose-dump: row M, padded stride (rows m and m+8 hit
            // disjoint bank groups; selection row reads are conflict-free)
#pragma unroll
            for (int v = 0; v < 8; ++v)
                DL[wave][t][(v + 8 * hi) * ROWS + n] = c[v];
        }
        // wave-private slab: DS ops of a wave complete in order; only need to
        // stop the compiler from reordering the cross-lane store->load.
        __builtin_amdgcn_wave_barrier();

        // ---- selection: lane (n,hi) owns query n, 16-pt tile hi ----
#pragma unroll
        for (int j = 0; j < 16; ++j) {
            float d  = DL[wave][hi][n * ROWS + j];
            int   id = base + hi * 16 + j;
            if (d < bd[KNN - 1]) {              // strict: ties keep lower idx
                bd[KNN - 1] = d; bi[KNN - 1] = id;
#pragma unroll
                for (int k = KNN - 1; k > 0; --k) {
                    if (bd[k] < bd[k - 1]) {    // strict: stable on ties
                        float td = bd[k]; bd[k] = bd[k - 1]; bd[k - 1] = td;
                        int   ti = bi[k]; bi[k] = bi[k - 1]; bi[k - 1] = ti;
                    }
                }
            }
        }
        // keep next iteration's DL stores behind this iteration's reads
        __builtin_amdgcn_wave_barrier();
    }

    // ---- dump 8 sorted candidate lists per query to LDS ----
    {
        int src = wave * 2 + hi;                // 0..7
#pragma unroll
        for (int k = 0; k < KNN; ++k) {
            MD[(n * 8 + src) * 16 + k] = bd[k];
            MI[(n * 8 + src) * 16 + k] = bi[k];
        }
    }
    __syncthreads();

    // ---- 8-way merge, 16 outputs per query, (dist, idx) lexicographic ----
    if (tid < 16) {
        int q = tid;
        int ptr[8];
#pragma unroll
        for (int s = 0; s < 8; ++s) ptr[s] = 0;

        long long* oi = out_idx + ((size_t)b * NQ + qbase + q) * KNN;

        for (int k = 0; k < KNN; ++k) {
            float best  = 3.9e38f;
            int   besti = 0x7fffffff;
            int   bs    = -1;
#pragma unroll
            for (int s = 0; s < 8; ++s) {
                if (ptr[s] < KNN) {
                    float d = MD[(q * 8 + s) * 16 + ptr[s]];
                    int   i = MI[(q * 8 + s) * 16 + ptr[s]];
                    if (d < best || (d == best && i < besti)) {
                        best = d; besti = i; bs = s;
                    }
                }
            }
#pragma unroll
            for (int s = 0; s < 8; ++s) if (s == bs) ptr[s]++;
            oi[k] = (long long)besti;
        }
    }
}

extern "C" void kernel_launch(void* const* d_in, const int* in_sizes, int n_in,
                              void* d_out, int out_size, void* d_ws, size_t ws_size,
                              hipStream_t stream) {
    (void)in_sizes; (void)n_in; (void)d_ws; (void)ws_size; (void)out_size;

    const float* xyz = (const float*)d_in[0];
    // Outputs concatenated flat in return order: idx (int64, B*NQ*K), then pts (f32, B*NQ*3).
    long long* out_idx = (long long*)d_out;
    float*     out_pts = (float*)((char*)d_out +
                                  (size_t)BATCH * NQ * KNN * sizeof(long long));

    dim3 grid(BATCH * (NQ / 16));
    dim3 block(TPB);
    knn_wmma_kernel<<<grid, block, 0, stream>>>(xyz, out_idx, out_pts);
}